// CCSolver_81406810128422
// MI455X (gfx1250) — compile-verified
//
#include <hip/hip_runtime.h>
#include <hip/hip_bf16.h>

// ---------------------------------------------------------------------------
// 8-connected CCL via two-pass lock-free union-find.
// parent array lives in d_out (int32, N elements); final pass rewrites it
// in-place into labels (root+1 for foreground, 0 for background).
// ---------------------------------------------------------------------------

#define CCL_W 2048          // reference fixed shape (1,1,2048,2048)
#define CCL_W_SHIFT 11
#define TILE 32

__device__ __forceinline__ int ld_relaxed_agent(const int* p) {
    return __hip_atomic_load(p, __ATOMIC_RELAXED, __HIP_MEMORY_SCOPE_AGENT);
}
__device__ __forceinline__ int ld_relaxed_wg(const int* p) {
    return __hip_atomic_load(p, __ATOMIC_RELAXED, __HIP_MEMORY_SCOPE_WORKGROUP);
}

// ---- global union-find -----------------------------------------------------
__device__ __forceinline__ int gfind(int* p, int i) {
    int j = ld_relaxed_agent(p + i);
    while (j != i) {
        int k = ld_relaxed_agent(p + j);
        if (k != j) atomicMin(&p[i], k);   // path halving; never raises pointer
        i = j;
        j = k;
    }
    return i;
}

__device__ __forceinline__ void gmerge(int* p, int a, int b) {
    while (true) {
        a = gfind(p, a);
        b = gfind(p, b);
        if (a == b) return;
        int lo = a < b ? a : b;
        int hi = a < b ? b : a;
        int old = atomicMin(&p[hi], lo);
        if (old == hi) return;             // linked a true root
        a = lo;
        b = old;                           // root moved underneath us; retry
    }
}

// ---- LDS (tile-local) union-find ------------------------------------------
__device__ __forceinline__ int lfind(int* p, int i) {
    int j = ld_relaxed_wg(p + i);
    while (j != i) {
        i = j;
        j = ld_relaxed_wg(p + i);
    }
    return i;
}

__device__ __forceinline__ void lmerge(int* p, int a, int b) {
    while (true) {
        a = lfind(p, a);
        b = lfind(p, b);
        if (a == b) return;
        int lo = a < b ? a : b;
        int hi = a < b ? b : a;
        int old = atomicMin(&p[hi], lo);   // ds_min_rtn_b32 on LDS
        if (old == hi) return;
        a = lo;
        b = old;
    }
}

// ---------------------------------------------------------------------------
// K1: threshold + tile-local CCL in LDS; emit global parent pointers.
// ---------------------------------------------------------------------------
__global__ __launch_bounds__(TILE * TILE)
void k_init_local(const float* __restrict__ prob, int* __restrict__ parent) {
    __shared__ int lab[TILE * TILE];

    const int lx = threadIdx.x;
    const int ly = threadIdx.y;
    const int gx = blockIdx.x * TILE + lx;
    const int gy = blockIdx.y * TILE + ly;
    const int gidx = (gy << CCL_W_SHIFT) + gx;
    const int lidx = ly * TILE + lx;

    const bool fg = prob[gidx] > 0.5f;
    lab[lidx] = fg ? lidx : -1;
    __syncthreads();

    if (fg) {
        if (lx > 0 && lab[lidx - 1] >= 0)
            lmerge(lab, lidx, lidx - 1);                  // W
        if (ly > 0) {
            if (lab[lidx - TILE] >= 0)
                lmerge(lab, lidx, lidx - TILE);           // N
            if (lx > 0 && lab[lidx - TILE - 1] >= 0)
                lmerge(lab, lidx, lidx - TILE - 1);       // NW
            if (lx < TILE - 1 && lab[lidx - TILE + 1] >= 0)
                lmerge(lab, lidx, lidx - TILE + 1);       // NE
        }
    }
    __syncthreads();

    int pv = gidx;                                        // background: self-root
    if (fg) {
        const int r = lfind(lab, lidx);                   // tree is static now
        const int rlx = r & (TILE - 1);
        const int rly = r >> 5;
        pv = ((blockIdx.y * TILE + rly) << CCL_W_SHIFT) + blockIdx.x * TILE + rlx;
    }
    parent[gidx] = pv;
}

// ---------------------------------------------------------------------------
// K2: union across tile borders only (~9% of pixels do work).
// ---------------------------------------------------------------------------
__global__ __launch_bounds__(256)
void k_border_merge(const float* __restrict__ prob, int* __restrict__ parent, int n) {
    const int gid = blockIdx.x * blockDim.x + threadIdx.x;
    if (gid >= n) return;

    const int gx = gid & (CCL_W - 1);
    const int gy = gid >> CCL_W_SHIFT;
    const int lx = gx & (TILE - 1);
    const int ly = gy & (TILE - 1);

    // interior pixels have all 4 predecessor edges inside their tile -> done in K1
    if (lx != 0 && ly != 0 && lx != TILE - 1) return;

    // warm the cross-row neighbor cacheline (gfx1250 global_prefetch_b8)
    if (gy > 0) __builtin_prefetch(prob + gid - CCL_W, 0, 0);

    if (!(prob[gid] > 0.5f)) return;

    if (lx == 0 && gx > 0 && prob[gid - 1] > 0.5f)
        gmerge(parent, gid, gid - 1);                     // W crosses tile
    if (gy > 0) {
        const int up = gid - CCL_W;
        if (ly == 0 && prob[up] > 0.5f)
            gmerge(parent, gid, up);                      // N crosses tile
        if ((lx == 0 || ly == 0) && gx > 0 && prob[up - 1] > 0.5f)
            gmerge(parent, gid, up - 1);                  // NW crosses tile
        if ((ly == 0 || lx == TILE - 1) && gx < CCL_W - 1 && prob[up + 1] > 0.5f)
            gmerge(parent, gid, up + 1);                  // NE crosses tile
    }
}

// ---------------------------------------------------------------------------
// K3: flatten every pointer to its root (idempotent, benign races).
// ---------------------------------------------------------------------------
__global__ __launch_bounds__(256)
void k_flatten(int* __restrict__ parent, int n) {
    const int i = blockIdx.x * blockDim.x + threadIdx.x;
    if (i >= n) return;
    const int r = gfind(parent, i);
    parent[i] = r;
}

// ---------------------------------------------------------------------------
// K4: labels = mask ? root+1 : 0   (in-place on d_out, x4 vectorized)
// ---------------------------------------------------------------------------
__global__ __launch_bounds__(256)
void k_finalize(const float4* __restrict__ prob4, int4* __restrict__ out4, int n4) {
    const int i = blockIdx.x * blockDim.x + threadIdx.x;
    if (i >= n4) return;
    const float4 p = prob4[i];
    int4 v = out4[i];
    v.x = (p.x > 0.5f) ? v.x + 1 : 0;
    v.y = (p.y > 0.5f) ? v.y + 1 : 0;
    v.z = (p.z > 0.5f) ? v.z + 1 : 0;
    v.w = (p.w > 0.5f) ? v.w + 1 : 0;
    out4[i] = v;
}

extern "C" void kernel_launch(void* const* d_in, const int* in_sizes, int n_in,
                              void* d_out, int out_size, void* d_ws, size_t ws_size,
                              hipStream_t stream) {
    (void)n_in; (void)d_ws; (void)ws_size; (void)out_size;

    const float* prob = (const float*)d_in[0];
    int* parent = (int*)d_out;                 // parent array == output buffer
    const int N = in_sizes[0];                 // 2048*2048
    const int H = N / CCL_W;

    // K1: per-tile local CCL (32x32 tiles, 1024 threads = 32 waves)
    dim3 b1(TILE, TILE);
    dim3 g1(CCL_W / TILE, H / TILE);
    k_init_local<<<g1, b1, 0, stream>>>(prob, parent);

    // K2: cross-tile unions
    const int T = 256;
    k_border_merge<<<(N + T - 1) / T, T, 0, stream>>>(prob, parent, N);

    // K3: flatten to roots
    k_flatten<<<(N + T - 1) / T, T, 0, stream>>>(parent, N);

    // K4: write final labels
    const int N4 = N / 4;
    k_finalize<<<(N4 + T - 1) / T, T, 0, stream>>>((const float4*)prob, (int4*)parent, N4);
}